// AGraphATLayer_56684978372724
// MI455X (gfx1250) — compile-verified
//
#include <hip/hip_runtime.h>

#define OUTF 96
#define NCOLT 6          // 96 / 16 column tiles
#define BN_EPS 1e-5f
#define LDS_STRIDE 98    // padded dword stride: conflict-free for stride-1 row access

typedef float v2f __attribute__((ext_vector_type(2)));
typedef float v8f __attribute__((ext_vector_type(8)));

#define WMMA_F32(a, b, c) \
    __builtin_amdgcn_wmma_f32_16x16x4_f32(false, (a), false, (b), (short)0, (c), false, false)

// ---------------------------------------------------------------------------
// Generic node GEMM: Y = act(X @ W^T + bias), B fragments from global (W is
// L1-resident). Used for the attention projection (K=64, no bias/relu).
// Fragment layouts per CDNA5 ISA 7.12.2:
//   A (16x4): lane L -> row = L&15; VGPR0 = K{0|2}, VGPR1 = K{1|3}
//   B (4x16): lane L -> col = L&15; same K striping
//   D (16x16): VGPR v -> row v / v+8 (half-wave), col = L&15
// ---------------------------------------------------------------------------
__global__ void node_gemm_wmma(const float* __restrict__ X,
                               const float* __restrict__ W,
                               const float* __restrict__ bias,
                               float* __restrict__ Y,
                               int N, int K, int relu_flag)
{
    const int lane       = threadIdx.x & 31;
    const int tile       = blockIdx.x * (blockDim.x >> 5) + (threadIdx.x >> 5);
    const int numTiles   = (N + 15) >> 4;
    if (tile >= numTiles) return;

    const int row0    = tile << 4;
    const int halfSel = lane >> 4;
    const int l15     = lane & 15;

    const int arow = min(row0 + l15, N - 1);
    const float* __restrict__ ap = X + (size_t)arow * K + (halfSel << 1);

    v8f acc[NCOLT];
#pragma unroll
    for (int t = 0; t < NCOLT; ++t) acc[t] = (v8f){0.f,0.f,0.f,0.f,0.f,0.f,0.f,0.f};

    for (int k0 = 0; k0 < K; k0 += 4) {
        v2f a;
        a.x = ap[k0];
        a.y = ap[k0 + 1];
#pragma unroll
        for (int t = 0; t < NCOLT; ++t) {
            const float* __restrict__ wp =
                W + (size_t)((t << 4) + l15) * K + (halfSel << 1) + k0;
            v2f b;
            b.x = wp[0];
            b.y = wp[1];
            acc[t] = WMMA_F32(a, b, acc[t]);
        }
    }

#pragma unroll
    for (int t = 0; t < NCOLT; ++t) {
        const int n    = (t << 4) + l15;
        const float bv = bias ? bias[n] : 0.f;
#pragma unroll
        for (int v = 0; v < 8; ++v) {
            const int m = row0 + v + (halfSel << 3);
            if (m < N) {
                float d = acc[t][v] + bv;
                if (relu_flag) d = fmaxf(d, 0.f);
                Y[(size_t)m * OUTF + n] = d;
            }
        }
    }
}

// ---------------------------------------------------------------------------
// Fused dual GEMM: Y0 = relu(X@W0^T + b0), Y1 = X@W1^T + b1.
// Shares A-fragment loads of X (feature read ONCE), and stages both weight
// matrices in LDS (2 x 96 x 98 dwords = 75 KB of the 320 KB WGP LDS) with a
// bank-conflict-free padded stride, so the WMMA loop is fed by ds_load_b64.
// ---------------------------------------------------------------------------
__global__ void node_gemm2_wmma(const float* __restrict__ X,   // [N x 96]
                                const float* __restrict__ W0,
                                const float* __restrict__ b0,
                                const float* __restrict__ W1,
                                const float* __restrict__ b1,
                                float* __restrict__ Y0,
                                float* __restrict__ Y1,
                                int N)
{
    extern __shared__ float sW[];   // [2][96][LDS_STRIDE]

    // Cooperative stage of both 96x96 weight matrices.
    for (int i = threadIdx.x; i < OUTF * OUTF; i += blockDim.x) {
        const int r = i / OUTF, c = i - r * OUTF;
        sW[r * LDS_STRIDE + c]                      = W0[i];
        sW[(OUTF + r) * LDS_STRIDE + c]             = W1[i];
    }
    __syncthreads();

    const int lane     = threadIdx.x & 31;
    const int tile     = blockIdx.x * (blockDim.x >> 5) + (threadIdx.x >> 5);
    const int numTiles = (N + 15) >> 4;
    if (tile >= numTiles) return;

    const int row0    = tile << 4;
    const int halfSel = lane >> 4;
    const int l15     = lane & 15;

    const int arow = min(row0 + l15, N - 1);
    const float* __restrict__ ap = X + (size_t)arow * OUTF + (halfSel << 1);

    v8f acc0[NCOLT], acc1[NCOLT];
#pragma unroll
    for (int t = 0; t < NCOLT; ++t) {
        acc0[t] = (v8f){0.f,0.f,0.f,0.f,0.f,0.f,0.f,0.f};
        acc1[t] = (v8f){0.f,0.f,0.f,0.f,0.f,0.f,0.f,0.f};
    }

    // Lane-constant part of the LDS fragment address (8-byte aligned: all
    // components even in dword units).
    const int wbase = l15 * LDS_STRIDE + (halfSel << 1);

    for (int k0 = 0; k0 < OUTF; k0 += 4) {
        v2f a;
        a.x = ap[k0];
        a.y = ap[k0 + 1];
#pragma unroll
        for (int t = 0; t < NCOLT; ++t) {
            const int widx = wbase + (t << 4) * LDS_STRIDE + k0;
            const v2f bA = *(const v2f*)&sW[widx];
            const v2f bB = *(const v2f*)&sW[OUTF * LDS_STRIDE + widx];
            acc0[t] = WMMA_F32(a, bA, acc0[t]);
            acc1[t] = WMMA_F32(a, bB, acc1[t]);
        }
    }

#pragma unroll
    for (int t = 0; t < NCOLT; ++t) {
        const int n    = (t << 4) + l15;
        const float b0v = b0[n];
        const float b1v = b1[n];
#pragma unroll
        for (int v = 0; v < 8; ++v) {
            const int m = row0 + v + (halfSel << 3);
            if (m < N) {
                Y0[(size_t)m * OUTF + n] = fmaxf(acc0[t][v] + b0v, 0.f);
                Y1[(size_t)m * OUTF + n] = acc1[t][v] + b1v;
            }
        }
    }
}

// ---------------------------------------------------------------------------
// Edge pass: one wave per edge; lane handles channels {lane, lane+32, lane+64}.
// msg = (A[src] - A[dst]) * C[src]; atomic scatter-add into H[dst].
// All node arrays (19.2 MB each) are L2-resident (192 MB) -> on-die traffic.
// ---------------------------------------------------------------------------
__global__ void edge_scatter(const float* __restrict__ Anode,
                             const float* __restrict__ Cnode,
                             const int* __restrict__ src,
                             const int* __restrict__ dst,
                             float* __restrict__ H, int E)
{
    const int gtid = blockIdx.x * blockDim.x + threadIdx.x;
    const int e    = gtid >> 5;
    const int lane = threadIdx.x & 31;
    if (e >= E) return;

    const int s = src[e];
    const int d = dst[e];
    const float* __restrict__ As = Anode + (size_t)s * OUTF;
    const float* __restrict__ Ad = Anode + (size_t)d * OUTF;
    const float* __restrict__ Cs = Cnode + (size_t)s * OUTF;
    float* __restrict__ Hd = H + (size_t)d * OUTF;

#pragma unroll
    for (int i = 0; i < 3; ++i) {
        const int c = lane + (i << 5);
        const float msg = (As[c] - Ad[c]) * Cs[c];
        (void)__hip_atomic_fetch_add(Hd + c, msg,
                                     __ATOMIC_RELAXED, __HIP_MEMORY_SCOPE_AGENT);
    }
}

// ---------------------------------------------------------------------------
// In-place ReLU + per-channel sum / sum-of-squares (thread c owns channel c,
// register accumulation, one atomic pair per channel per block).
// ---------------------------------------------------------------------------
__global__ void relu_stats(float* __restrict__ H, float* __restrict__ stats, int N)
{
    const int c = threadIdx.x;  // 0..95
    float s = 0.f, s2 = 0.f;
    for (int r = blockIdx.x; r < N; r += gridDim.x) {
        const size_t idx = (size_t)r * OUTF + c;
        float v = H[idx];
        v = fmaxf(v, 0.f);
        H[idx] = v;
        s  += v;
        s2 += v * v;
    }
    (void)__hip_atomic_fetch_add(&stats[c], s,
                                 __ATOMIC_RELAXED, __HIP_MEMORY_SCOPE_AGENT);
    (void)__hip_atomic_fetch_add(&stats[OUTF + c], s2,
                                 __ATOMIC_RELAXED, __HIP_MEMORY_SCOPE_AGENT);
}

__global__ void bn_finalize(float* __restrict__ stats,
                            const float* __restrict__ gamma,
                            const float* __restrict__ beta, int N)
{
    const int c = threadIdx.x;
    if (c < OUTF) {
        const float inv  = 1.0f / (float)N;
        const float mean = stats[c] * inv;
        const float var  = stats[OUTF + c] * inv - mean * mean;
        const float sc   = gamma[c] * rsqrtf(var + BN_EPS);
        stats[2 * OUTF + c] = sc;
        stats[3 * OUTF + c] = beta[c] - mean * sc;
    }
}

__global__ void bn_apply(float* __restrict__ H, const float* __restrict__ stats,
                         long total)
{
    const long i = (long)blockIdx.x * blockDim.x + threadIdx.x;
    if (i < total) {
        const int c = (int)(i % OUTF);
        H[i] = H[i] * stats[2 * OUTF + c] + stats[3 * OUTF + c];
    }
}

// ---------------------------------------------------------------------------
// Launch.  Inputs (setup_inputs order):
//  0 feature[N,96] 1 sp_embeddings[N,64] 2 src[E] 3 dst[E]
//  4 W_conv[96,96] 5 b_conv[96] 6 W_self[96,96] 7 b_self[96]
//  8 W_att[96,64]  9 gamma[96] 10 beta[96]
// ---------------------------------------------------------------------------
extern "C" void kernel_launch(void* const* d_in, const int* in_sizes, int n_in,
                              void* d_out, int out_size, void* d_ws, size_t ws_size,
                              hipStream_t stream)
{
    const float* feature = (const float*)d_in[0];
    const float* sp_emb  = (const float*)d_in[1];
    const int*   src     = (const int*)d_in[2];
    const int*   dst     = (const int*)d_in[3];
    const float* W_conv  = (const float*)d_in[4];
    const float* b_conv  = (const float*)d_in[5];
    const float* W_self  = (const float*)d_in[6];
    const float* b_self  = (const float*)d_in[7];
    const float* W_att   = (const float*)d_in[8];
    const float* gamma   = (const float*)d_in[9];
    const float* beta    = (const float*)d_in[10];

    const int N     = in_sizes[0] / OUTF;  // 50000
    const int E     = in_sizes[2];         // 800000
    const int EMBED = in_sizes[1] / N;     // 64

    float* Anode = (float*)d_ws;                 // [N x 96] attention proj
    float* Cnode = Anode + (size_t)N * OUTF;     // [N x 96] relu conv
    float* stats = Cnode + (size_t)N * OUTF;     // [4 x 96]
    float* H     = (float*)d_out;                // accumulator / output

    hipMemsetAsync(stats, 0, 2 * OUTF * sizeof(float), stream);

    const int numTiles = (N + 15) / 16;

    // Attention projection (K=64): generic WMMA GEMM, 4 waves/block.
    node_gemm_wmma<<<(numTiles + 3) / 4, 128, 0, stream>>>(
        sp_emb, W_att, nullptr, Anode, N, EMBED, 0);

    // Fused conv(+relu) and self GEMMs: feature read once, W staged in LDS.
    const size_t ldsBytes = (size_t)2 * OUTF * LDS_STRIDE * sizeof(float);
    node_gemm2_wmma<<<(numTiles + 7) / 8, 256, ldsBytes, stream>>>(
        feature, W_conv, b_conv, W_self, b_self, Cnode, H, N);

    // Edge scatter: one wave per edge, 8 waves per 256-thread block.
    edge_scatter<<<(E + 7) / 8, 256, 0, stream>>>(Anode, Cnode, src, dst, H, E);

    // ReLU + batch stats, BN fold, BN apply.
    relu_stats<<<512, OUTF, 0, stream>>>(H, stats, N);
    bn_finalize<<<1, OUTF, 0, stream>>>(stats, gamma, beta, N);

    const long total = (long)N * OUTF;
    bn_apply<<<(int)((total + 255) / 256), 256, 0, stream>>>(H, stats, total);
}